// GAT_82197084110909
// MI455X (gfx1250) — compile-verified
//
#include <hip/hip_runtime.h>
#include <hip/hip_bf16.h>

typedef __attribute__((ext_vector_type(2))) float v2f;
typedef __attribute__((ext_vector_type(8))) float v8f;

#define GAT_N 100000
#define GAT_E 1600000
#define NEG_SLOPE 0.2f

// ---------- order-preserving float <-> uint mapping for atomicMax ----------
__device__ __forceinline__ unsigned f2ord(float f) {
    unsigned u = __float_as_uint(f);
    return (u & 0x80000000u) ? ~u : (u | 0x80000000u);
}
__device__ __forceinline__ float ord2f(unsigned u) {
    return (u & 0x80000000u) ? __uint_as_float(u & 0x7fffffffu)
                             : __uint_as_float(~u);
}
__device__ __forceinline__ float lrelu(float v) {
    return (v >= 0.f) ? v : NEG_SLOPE * v;
}

// ---------- WMMA f32 GEMM: C[n,NC] = A[n,K] * B[K,NC] ----------
// One wave per 16x16 output tile; K fully unrolled (straight-line WMMA chain).
// A-frag (ISA 7.12.2, 32-bit A 16x4): lanes 0-15 -> K={k,k+1},
// lanes 16-31 -> K={k+2,k+3}, M = lane%16. B mirrors with N = lane%16.
// C/D: VGPR v -> row rowBase + v + 8*(lane>=16), col = colBase + lane%16.
template <int K, int NC>
__global__ __launch_bounds__(32)
void wmma_gemm_f32(const float* __restrict__ A, const float* __restrict__ B,
                   float* __restrict__ C) {
    const int rowBase = blockIdx.x * 16;
    const int colBase = blockIdx.y * 16;
    const int lane = threadIdx.x;
    const int half = lane >> 4;
    const int l16  = lane & 15;

    v8f acc = {};
    const float* Arow = A + (size_t)(rowBase + l16) * K;
#pragma unroll
    for (int k = 0; k < K; k += 4) {
        const int kk = k + 2 * half;
        v2f a;
        a.x = Arow[kk];
        a.y = Arow[kk + 1];
        v2f b;
        b.x = B[(size_t)kk * NC + colBase + l16];
        b.y = B[(size_t)(kk + 1) * NC + colBase + l16];
        acc = __builtin_amdgcn_wmma_f32_16x16x4_f32(
            false, a, false, b, (short)0, acc, false, false);
    }
#pragma unroll
    for (int v = 0; v < 8; ++v) {
        const int row = rowBase + v + 8 * half;
        C[(size_t)row * NC + colBase + l16] = acc[v];
    }
}

// ---------- per-(node, head) attention scores el/er ----------
template <int H, int D>
__global__ void score_kernel(const float* __restrict__ feat,
                             const float* __restrict__ al,
                             const float* __restrict__ ar,
                             float* __restrict__ el, float* __restrict__ er,
                             int n) {
    const int i = blockIdx.x * blockDim.x + threadIdx.x;
    if (i >= n * H) return;
    const int node = i / H, h = i % H;          // H = power of 2 -> shifts
    const float* f = feat + (size_t)node * (H * D) + h * D;
    float sl = 0.f, sr = 0.f;
#pragma unroll
    for (int d = 0; d < D; d += 4) {
        const float4 v = *(const float4*)(f + d);
        const float4 wl = *(const float4*)(al + h * D + d);
        const float4 wr = *(const float4*)(ar + h * D + d);
        sl += v.x * wl.x + v.y * wl.y + v.z * wl.z + v.w * wl.w;
        sr += v.x * wr.x + v.y * wr.y + v.z * wr.z + v.w * wr.w;
    }
    el[i] = sl;
    er[i] = sr;
}

// ---------- edge pass A: e = leaky_relu(el[src]+er[dst]); segment max ----------
// One thread per edge; all H heads handled with vector loads.
template <int H>
__global__ void edge_passA(const int* __restrict__ src, const int* __restrict__ dst,
                           const float* __restrict__ el, const float* __restrict__ er,
                           float* __restrict__ ebuf, unsigned* __restrict__ emax,
                           int nE) {
    const int e = blockIdx.x * blockDim.x + threadIdx.x;
    if (e >= nE) return;
    const int s = src[e], d = dst[e];
    if constexpr (H == 4) {
        const float4 a = *(const float4*)(el + s * 4);
        const float4 b = *(const float4*)(er + d * 4);
        float4 v;
        v.x = lrelu(a.x + b.x);
        v.y = lrelu(a.y + b.y);
        v.z = lrelu(a.z + b.z);
        v.w = lrelu(a.w + b.w);
        *(float4*)(ebuf + (size_t)e * 4) = v;
        atomicMax(&emax[d * 4 + 0], f2ord(v.x));
        atomicMax(&emax[d * 4 + 1], f2ord(v.y));
        atomicMax(&emax[d * 4 + 2], f2ord(v.z));
        atomicMax(&emax[d * 4 + 3], f2ord(v.w));
    } else {
#pragma unroll
        for (int h = 0; h < H; ++h) {
            const float v = lrelu(el[s * H + h] + er[d * H + h]);
            ebuf[(size_t)e * H + h] = v;
            atomicMax(&emax[d * H + h], f2ord(v));
        }
    }
}

// ---------- edge pass B: ex = exp(e - max); segment sum ----------
template <int H>
__global__ void edge_passB(const int* __restrict__ dst,
                           float* __restrict__ ebuf,
                           const unsigned* __restrict__ emax,
                           float* __restrict__ denom, int nE) {
    const int e = blockIdx.x * blockDim.x + threadIdx.x;
    if (e >= nE) return;
    const int d = dst[e];
    if constexpr (H == 4) {
        float4 v = *(const float4*)(ebuf + (size_t)e * 4);
        v.x = __expf(v.x - ord2f(emax[d * 4 + 0]));
        v.y = __expf(v.y - ord2f(emax[d * 4 + 1]));
        v.z = __expf(v.z - ord2f(emax[d * 4 + 2]));
        v.w = __expf(v.w - ord2f(emax[d * 4 + 3]));
        *(float4*)(ebuf + (size_t)e * 4) = v;
        atomicAdd(&denom[d * 4 + 0], v.x);
        atomicAdd(&denom[d * 4 + 1], v.y);
        atomicAdd(&denom[d * 4 + 2], v.z);
        atomicAdd(&denom[d * 4 + 3], v.w);
    } else {
#pragma unroll
        for (int h = 0; h < H; ++h) {
            const float ex = __expf(ebuf[(size_t)e * H + h] - ord2f(emax[d * H + h]));
            ebuf[(size_t)e * H + h] = ex;
            atomicAdd(&denom[d * H + h], ex);
        }
    }
}

// ---------- edge aggregate: out[dst] += feat[src] * alpha ----------
// One thread per (edge, head, 4-wide d-group): b128 gather of feat[src],
// 4 atomics into one out cacheline. A full wave covers one edge's
// contiguous H*D row when H*D/4 == 32 (layer 1) -> perfectly coalesced.
template <int H, int D>
__global__ void edge_aggregate(const int* __restrict__ src, const int* __restrict__ dst,
                               const float* __restrict__ ebuf,
                               const float* __restrict__ denom,
                               const float* __restrict__ feat,
                               float* __restrict__ out, int nE) {
    constexpr int G  = D / 4;    // 4-wide groups per head
    constexpr int GH = G * H;    // groups per edge (power of 2)
    const long long i = (long long)blockIdx.x * blockDim.x + threadIdx.x;
    if (i >= (long long)nE * GH) return;
    const int e  = (int)(i / GH);           // constexpr pow2 -> shifts
    const int r  = (int)(i % GH);
    const int h  = r / G;
    const int dd = (r % G) * 4;
    const int s = src[e], t = dst[e];
    const float alpha = ebuf[(size_t)e * H + h] / denom[t * H + h];
    const float4 f = *(const float4*)(feat + (size_t)s * (H * D) + h * D + dd);
    float* o = out + (size_t)t * (H * D) + h * D + dd;
    atomicAdd(o + 0, f.x * alpha);
    atomicAdd(o + 1, f.y * alpha);
    atomicAdd(o + 2, f.z * alpha);
    atomicAdd(o + 3, f.w * alpha);
}

// ---------- layer-1 epilogue: h = relu(mean_h(out + b1)) ----------
template <int H, int D>
__global__ void finalize1(const float* __restrict__ out1, const float* __restrict__ b1,
                          float* __restrict__ hbuf, int n) {
    const int i = blockIdx.x * blockDim.x + threadIdx.x;
    if (i >= n * D) return;
    const int node = i / D, d = i % D;
    float s = 0.f;
#pragma unroll
    for (int h = 0; h < H; ++h)
        s += out1[(size_t)node * (H * D) + h * D + d] + b1[h * D + d];
    s *= (1.0f / H);
    hbuf[i] = (s > 0.f) ? s : 0.f;
}

// ---------- layer-2 epilogue: d_out = out2 + b2 ----------
template <int D>
__global__ void finalize2(const float* __restrict__ out2, const float* __restrict__ b2,
                          float* __restrict__ dout, int n) {
    const int i = blockIdx.x * blockDim.x + threadIdx.x;
    if (i >= n * D) return;
    dout[i] = out2[i] + b2[i % D];
}

extern "C" void kernel_launch(void* const* d_in, const int* in_sizes, int n_in,
                              void* d_out, int out_size, void* d_ws, size_t ws_size,
                              hipStream_t stream) {
    (void)in_sizes; (void)n_in; (void)out_size; (void)ws_size;
    const float* x   = (const float*)d_in[0];
    const int*   src = (const int*)d_in[1];
    const int*   dst = (const int*)d_in[2];
    const float* W1  = (const float*)d_in[3];
    const float* al1 = (const float*)d_in[4];
    const float* ar1 = (const float*)d_in[5];
    const float* b1  = (const float*)d_in[6];
    const float* W2  = (const float*)d_in[7];
    const float* al2 = (const float*)d_in[8];
    const float* ar2 = (const float*)d_in[9];
    const float* b2  = (const float*)d_in[10];

    constexpr int Nn = GAT_N, Ee = GAT_E;
    constexpr int H1 = 4, D1 = 32, F1 = 128;   // layer 1: heads=4, hid=32, in=128
    constexpr int D2 = 16, K2 = 32;            // layer 2: heads=1, out=16, in=32

    // workspace carve-out (floats); all offsets multiples of 16B
    float* w = (float*)d_ws;
    float*    feat1  = w;            w += (size_t)Nn * F1;   // 51.2 MB (L2-resident)
    float*    out1   = w;            w += (size_t)Nn * F1;   // 51.2 MB (L2-resident)
    float*    ebuf   = w;            w += (size_t)Ee * H1;   // 25.6 MB
    float*    el1    = w;            w += (size_t)Nn * H1;
    float*    er1    = w;            w += (size_t)Nn * H1;
    unsigned* emax1  = (unsigned*)w; w += (size_t)Nn * H1;
    float*    denom1 = w;            w += (size_t)Nn * H1;
    float*    hbuf   = w;            w += (size_t)Nn * K2;
    float*    feat2  = w;            w += (size_t)Nn * D2;
    float*    out2   = w;            w += (size_t)Nn * D2;
    float*    el2    = w;            w += (size_t)Nn;
    float*    er2    = w;            w += (size_t)Nn;
    unsigned* emax2  = (unsigned*)w; w += (size_t)Nn;
    float*    denom2 = w;            w += (size_t)Nn;

    const int TB = 256;

    // zero-init accumulators (capture-safe async memsets);
    // 0 < f2ord(any finite) so 0 is a valid "-inf" for the ordered-max buffers
    hipMemsetAsync(emax1,  0, (size_t)Nn * H1 * 4, stream);
    hipMemsetAsync(denom1, 0, (size_t)Nn * H1 * 4, stream);
    hipMemsetAsync(out1,   0, (size_t)Nn * F1 * 4, stream);
    hipMemsetAsync(emax2,  0, (size_t)Nn * 4, stream);
    hipMemsetAsync(denom2, 0, (size_t)Nn * 4, stream);
    hipMemsetAsync(out2,   0, (size_t)Nn * D2 * 4, stream);

    // ===== layer 1 =====
    wmma_gemm_f32<F1, F1><<<dim3(Nn / 16, F1 / 16), 32, 0, stream>>>(x, W1, feat1);
    score_kernel<H1, D1><<<(Nn * H1 + TB - 1) / TB, TB, 0, stream>>>(feat1, al1, ar1, el1, er1, Nn);
    edge_passA<H1><<<(Ee + TB - 1) / TB, TB, 0, stream>>>(src, dst, el1, er1, ebuf, emax1, Ee);
    edge_passB<H1><<<(Ee + TB - 1) / TB, TB, 0, stream>>>(dst, ebuf, emax1, denom1, Ee);
    {
        const long long total = (long long)Ee * (H1 * D1 / 4);  // 51.2M threads
        edge_aggregate<H1, D1><<<(unsigned)((total + TB - 1) / TB), TB, 0, stream>>>(
            src, dst, ebuf, denom1, feat1, out1, Ee);
    }
    finalize1<H1, D1><<<(Nn * D1 + TB - 1) / TB, TB, 0, stream>>>(out1, b1, hbuf, Nn);

    // ===== layer 2 =====
    wmma_gemm_f32<K2, D2><<<dim3(Nn / 16, D2 / 16), 32, 0, stream>>>(hbuf, W2, feat2);
    score_kernel<1, D2><<<(Nn + TB - 1) / TB, TB, 0, stream>>>(feat2, al2, ar2, el2, er2, Nn);
    edge_passA<1><<<(Ee + TB - 1) / TB, TB, 0, stream>>>(src, dst, el2, er2, ebuf, emax2, Ee);
    edge_passB<1><<<(Ee + TB - 1) / TB, TB, 0, stream>>>(dst, ebuf, emax2, denom2, Ee);
    {
        const long long total = (long long)Ee * (D2 / 4);       // 6.4M threads
        edge_aggregate<1, D2><<<(unsigned)((total + TB - 1) / TB), TB, 0, stream>>>(
            src, dst, ebuf, denom2, feat2, out2, Ee);
    }
    finalize2<D2><<<(Nn * D2 + TB - 1) / TB, TB, 0, stream>>>(out2, b2, (float*)d_out, Nn);
}